// Generator_15899968930201
// MI455X (gfx1250) — compile-verified
//
#include <hip/hip_runtime.h>
#include <hip/hip_bf16.h>
#include <math.h>

typedef __attribute__((ext_vector_type(2))) float v2f;
typedef __attribute__((ext_vector_type(8))) float v8f;

// ---------------------------------------------------------------------------
// Kernel A: logits[bt, v] = x[bt,:] . proj_w[v,:] + proj_b[v]
// M = BT (1024), N = V (32000), K = H (512)
// Block: 256 threads = 8 waves. Block tile 64(M) x 256(N).
// Wave tile: 32(M) x 64(N) -> 2 A-frags x 4 B-frags = 8 WMMAs per K-step,
// fed by 6 b64 loads (wmma:load = 1.33). Software pipelined 2 deep.
// Addressing: one 64-bit pointer per stream, incremented by a constant per
// iteration; loads use immediate offsets. Unroll-by-2 so the double-buffer
// rotation is resolved by register renaming, not v_mov copies.
// ---------------------------------------------------------------------------

struct Frag6 {
    v2f a0, a1, b0, b1, b2, b3;
};

__global__ __launch_bounds__(256) void pg_gemm_logits(
    const float* __restrict__ x,     // [M, K] row-major
    const float* __restrict__ w,     // [N, K] row-major
    const float* __restrict__ bias,  // [N]
    float* __restrict__ logits,      // [M, N] row-major
    int M, int N, int K)
{
    const int tid  = threadIdx.x;
    const int wave = tid >> 5;
    const int lane = tid & 31;
    const int hh   = lane >> 4;   // half of the wave (K-split within frag)
    const int l    = lane & 15;   // M index for A frags, N index for B frags

    const int wm = wave & 1;      // 2 waves along M (32 rows each)
    const int wn = wave >> 1;     // 4 waves along N (64 cols each)
    const int m0 = blockIdx.y * 64  + wm * 32;
    const int n0 = blockIdx.x * 256 + wn * 64;

    const int kbase = 2 * hh;     // ISA layout: VGPR0 holds K=2*half, VGPR1 K=2*half+1

    // Per-lane stream pointers (pre-offset by kbase); bumped by +8 floats per
    // loop iteration so loads keep constant immediate offsets.
    const float* pa0 = x + (size_t)(m0 +  0 + l) * K + kbase;
    const float* pa1 = x + (size_t)(m0 + 16 + l) * K + kbase;
    const float* pb0 = w + (size_t)(n0 +  0 + l) * K + kbase;
    const float* pb1 = w + (size_t)(n0 + 16 + l) * K + kbase;
    const float* pb2 = w + (size_t)(n0 + 32 + l) * K + kbase;
    const float* pb3 = w + (size_t)(n0 + 48 + l) * K + kbase;

    auto ldf = [&](int ko) -> Frag6 {
        Frag6 f;
        f.a0 = *(const v2f*)(pa0 + ko);
        f.a1 = *(const v2f*)(pa1 + ko);
        f.b0 = *(const v2f*)(pb0 + ko);
        f.b1 = *(const v2f*)(pb1 + ko);
        f.b2 = *(const v2f*)(pb2 + ko);
        f.b3 = *(const v2f*)(pb3 + ko);
        return f;
    };

    v8f acc00 = {}, acc01 = {}, acc02 = {}, acc03 = {};
    v8f acc10 = {}, acc11 = {}, acc12 = {}, acc13 = {};

    #define WMMA_STEP(f)                                                                              \
        do {                                                                                          \
            acc00 = __builtin_amdgcn_wmma_f32_16x16x4_f32(false, (f).a0, false, (f).b0, (short)0, acc00, false, false); \
            acc01 = __builtin_amdgcn_wmma_f32_16x16x4_f32(false, (f).a0, false, (f).b1, (short)0, acc01, false, false); \
            acc02 = __builtin_amdgcn_wmma_f32_16x16x4_f32(false, (f).a0, false, (f).b2, (short)0, acc02, false, false); \
            acc03 = __builtin_amdgcn_wmma_f32_16x16x4_f32(false, (f).a0, false, (f).b3, (short)0, acc03, false, false); \
            acc10 = __builtin_amdgcn_wmma_f32_16x16x4_f32(false, (f).a1, false, (f).b0, (short)0, acc10, false, false); \
            acc11 = __builtin_amdgcn_wmma_f32_16x16x4_f32(false, (f).a1, false, (f).b1, (short)0, acc11, false, false); \
            acc12 = __builtin_amdgcn_wmma_f32_16x16x4_f32(false, (f).a1, false, (f).b2, (short)0, acc12, false, false); \
            acc13 = __builtin_amdgcn_wmma_f32_16x16x4_f32(false, (f).a1, false, (f).b3, (short)0, acc13, false, false); \
        } while (0)

    // Prologue: fragments for k = 0 and k = 4 in flight.
    Frag6 buf0 = ldf(0);
    Frag6 buf1 = ldf(4);

    // Steady state: unconditional prefetch at fixed immediate offsets;
    // pointers advance once per iteration. Peeled drain below.
    int k = 0;
    #pragma unroll 2
    for (; k + 16 <= K; k += 8) {
        Frag6 nxt0 = ldf(8);
        WMMA_STEP(buf0);          // waits on buf0's older loads, not the fresh prefetch
        Frag6 nxt1 = ldf(12);
        WMMA_STEP(buf1);
        pa0 += 8; pa1 += 8; pb0 += 8; pb1 += 8; pb2 += 8; pb3 += 8;
        buf0 = nxt0;
        buf1 = nxt1;
    }
    // Drain: last two K-steps, no prefetch.
    WMMA_STEP(buf0);
    WMMA_STEP(buf1);
    #undef WMMA_STEP

    // C/D layout: VGPR r holds D[M = r + 8*half][N = l]
    #pragma unroll
    for (int t = 0; t < 4; ++t) {
        const int n = n0 + t * 16 + l;
        const float bv = bias[n];
        const v8f* am0 = (t == 0) ? &acc00 : (t == 1) ? &acc01 : (t == 2) ? &acc02 : &acc03;
        const v8f* am1 = (t == 0) ? &acc10 : (t == 1) ? &acc11 : (t == 2) ? &acc12 : &acc13;
        #pragma unroll
        for (int r = 0; r < 8; ++r) {
            logits[(size_t)(m0 +  0 + r + 8 * hh) * N + n] = (*am0)[r] + bv;
            logits[(size_t)(m0 + 16 + r + 8 * hh) * N + n] = (*am1)[r] + bv;
        }
    }
}

// ---------------------------------------------------------------------------
// Kernel B: fused epilogue, one workgroup per (b,t) row.
// Stages the full 32050-float extended-vocab row in LDS (MI455X: 320KB/WGP),
// does both softmaxes, the LDS-atomic scatter, and the final log in one pass.
// Logits are re-read once for the exp pass -- they live in the 192MB L2.
// ---------------------------------------------------------------------------
#define NT 512           // threads per block
#define MAX_EXT 32064    // >= V + OOV (32050), padded

__device__ __forceinline__ float blk_reduce_max(float v, float* red, int tid) {
    red[tid] = v; __syncthreads();
    for (int s = NT >> 1; s > 0; s >>= 1) {
        if (tid < s) red[tid] = fmaxf(red[tid], red[tid + s]);
        __syncthreads();
    }
    float r = red[0]; __syncthreads();
    return r;
}

__device__ __forceinline__ float blk_reduce_sum(float v, float* red, int tid) {
    red[tid] = v; __syncthreads();
    for (int s = NT >> 1; s > 0; s >>= 1) {
        if (tid < s) red[tid] += red[tid + s];
        __syncthreads();
    }
    float r = red[0]; __syncthreads();
    return r;
}

__global__ __launch_bounds__(NT) void pg_finalize(
    const float* __restrict__ x,          // [BT, H]
    const float* __restrict__ attn_dist,  // [BT, S]
    const int*   __restrict__ enc_idx,    // [B, S]
    const float* __restrict__ pgen_w,     // [H]
    const float* __restrict__ pgen_b,     // [1]
    const float* __restrict__ logits,     // [BT, V]
    float* __restrict__ out,              // [BT, V+OOV]
    int T, int S, int H, int V, int OOV)
{
    __shared__ float ext[MAX_EXT];
    __shared__ float red[NT];

    const int bt  = blockIdx.x;
    const int b   = bt / T;
    const int tid = threadIdx.x;
    const int NE  = V + OOV;

    // ---- alpha = sigmoid(x . pgen_w + pgen_b) ----
    float p = 0.f;
    for (int h = tid; h < H; h += NT)
        p += x[(size_t)bt * H + h] * pgen_w[h];
    const float dot = blk_reduce_sum(p, red, tid);
    const float alpha = 1.f / (1.f + __expf(-(dot + pgen_b[0])));

    // ---- attention softmax statistics ----
    const float* arow = attn_dist + (size_t)bt * S;
    float amax = -INFINITY;
    for (int s = tid; s < S; s += NT) amax = fmaxf(amax, arow[s]);
    amax = blk_reduce_max(amax, red, tid);
    float asum = 0.f;
    for (int s = tid; s < S; s += NT) asum += __expf(arow[s] - amax);
    asum = blk_reduce_sum(asum, red, tid);
    const float ascale = (1.f - alpha) / asum;

    // ---- vocab softmax into LDS ----
    const float* lrow = logits + (size_t)bt * V;
    float lmax = -INFINITY;
    for (int j = tid; j < V; j += NT) lmax = fmaxf(lmax, lrow[j]);
    lmax = blk_reduce_max(lmax, red, tid);
    float lsum = 0.f;
    for (int j = tid; j < V; j += NT) {
        float e = __expf(lrow[j] - lmax);   // second read hits L2
        ext[j] = e;
        lsum += e;
    }
    lsum = blk_reduce_sum(lsum, red, tid);
    const float vscale = alpha / lsum;
    for (int j = tid; j < V; j += NT) ext[j] *= vscale;
    for (int j = V + tid; j < NE; j += NT) ext[j] = 0.f;  // OOV slots
    __syncthreads();

    // ---- scatter attention mass (ds_add_f32 LDS atomics) ----
    for (int s = tid; s < S; s += NT) {
        const float aval = __expf(arow[s] - amax) * ascale;
        int idx = enc_idx[(size_t)b * S + s];
        idx = idx < 0 ? 0 : (idx >= NE ? NE - 1 : idx);
        atomicAdd(&ext[idx], aval);
    }
    __syncthreads();

    // ---- final log, streamed out ----
    float* orow = out + (size_t)bt * NE;
    for (int j = tid; j < NE; j += NT)
        orow[j] = logf(ext[j] + 1e-18f);
}

// ---------------------------------------------------------------------------
extern "C" void kernel_launch(void* const* d_in, const int* in_sizes, int n_in,
                              void* d_out, int out_size, void* d_ws, size_t ws_size,
                              hipStream_t stream) {
    const float* x         = (const float*)d_in[0];  // [B,T,H]
    const float* attn_dist = (const float*)d_in[1];  // [B,T,S]
    const int*   enc_idx   = (const int*)  d_in[2];  // [B,S]
    const float* proj_w    = (const float*)d_in[3];  // [V,H]
    const float* proj_b    = (const float*)d_in[4];  // [V]
    const float* pgen_w    = (const float*)d_in[5];  // [1,H]
    const float* pgen_b    = (const float*)d_in[6];  // [1]
    float*       out       = (float*)d_out;

    const int H   = in_sizes[5];                 // 512
    const int V   = in_sizes[4];                 // 32000
    const int BT  = in_sizes[0] / H;             // 1024
    const int S   = in_sizes[1] / BT;            // 512
    const int NE  = out_size / BT;               // 32050
    const int OOV = NE - V;                      // 50
    const int Bb  = in_sizes[2] / S;             // 8
    const int T   = BT / Bb;                     // 128

    float* logits = (float*)d_ws;                // [BT, V] fp32, 131 MB (fits L2)

    dim3 gA(V / 256, BT / 64);                   // 125 x 16 workgroups
    pg_gemm_logits<<<gA, 256, 0, stream>>>(x, proj_w, proj_b, logits, BT, V, H);

    pg_finalize<<<BT, NT, 0, stream>>>(x, attn_dist, enc_idx, pgen_w, pgen_b,
                                       logits, out, T, S, H, V, OOV);
}